// GraphConv_18330920419888
// MI455X (gfx1250) — compile-verified
//
#include <hip/hip_runtime.h>
#include <stdint.h>
#include <stddef.h>

// ---------------- problem constants ----------------
#define N_USERS 50000
#define N_ITEMS 50000
#define NNODE   100000              // N_USERS + N_ITEMS
#define NNZE    3200000             // edges
#define DDIM    64
#define HALF_E  1600000             // NNZE/2   (threefry pairing for edge uniforms)
#define NDELEM  6400000             // NNODE*DDIM
#define HALF_M  3200000             // NDELEM/2 (threefry pairing for mess dropout)

// ---------------- threefry2x32 (matches JAX) ----------------
__host__ __device__ inline uint32_t rotl32(uint32_t x, int r) {
    return (x << r) | (x >> (32 - r));
}

__host__ __device__ inline void threefry2x32(uint32_t k0, uint32_t k1,
                                             uint32_t c0, uint32_t c1,
                                             uint32_t& o0, uint32_t& o1) {
    const uint32_t ks2 = 0x1BD11BDAu ^ k0 ^ k1;
    uint32_t x0 = c0 + k0, x1 = c1 + k1;
#define TFR(r) { x0 += x1; x1 = rotl32(x1, r); x1 ^= x0; }
    TFR(13) TFR(15) TFR(26) TFR(6)   x0 += k1;  x1 += ks2 + 1u;
    TFR(17) TFR(29) TFR(16) TFR(24)  x0 += ks2; x1 += k0  + 2u;
    TFR(13) TFR(15) TFR(26) TFR(6)   x0 += k0;  x1 += k1  + 3u;
    TFR(17) TFR(29) TFR(16) TFR(24)  x0 += k1;  x1 += ks2 + 4u;
    TFR(13) TFR(15) TFR(26) TFR(6)   x0 += ks2; x1 += k0  + 5u;
#undef TFR
    o0 = x0; o1 = x1;
}

__device__ inline float u01_from_bits(uint32_t b) {
    // jax.random.uniform: (bits>>9)|0x3f800000 bitcast -> [1,2) ; minus 1
    return __uint_as_float((b >> 9) | 0x3f800000u) - 1.0f;
}

// uniform for edge e of an array of NNZE draws (jax counts = iota(NNZE))
__device__ inline float edge_u(uint32_t k0, uint32_t k1, uint32_t e) {
    uint32_t o0, o1;
    const uint32_t p = (e < HALF_E) ? e : (e - HALF_E);
    threefry2x32(k0, k1, p, p + HALF_E, o0, o1);
    return u01_from_bits((e < HALF_E) ? o0 : o1);
}

// ---------------- gfx1250-specific instruction helpers ----------------
__device__ inline void async_g2l_b32(const void* gptr, void* lds_generic) {
    // VDST = LDS byte offset (low 32 bits of generic LDS pointer),
    // VADDR = 64-bit global address, no SADDR ("off").
    const uint32_t loff = (uint32_t)(uintptr_t)lds_generic;
    const uint64_t ga   = (uint64_t)(uintptr_t)gptr;
    asm volatile("global_load_async_to_lds_b32 %0, %1, off"
                 :: "v"(loff), "v"(ga) : "memory");
}

__device__ inline void atomic_add_f32_dev(float* p, float v) {
    // Guaranteed single hw FP atomic at device scope (performed in L2).
    // No "memory" clobber: volatile keeps ordering/branch placement, while the
    // compiler stays free to pipeline independent gather loads across it.
    const uint64_t ga = (uint64_t)(uintptr_t)p;
    asm volatile("global_atomic_add_f32 %0, %1, off scope:SCOPE_DEV"
                 :: "v"(ga), "v"(v));
}

// ---------------- kernel 1: build all_embed, write hop-0 slot ----------------
__global__ __launch_bounds__(256) void init_embed(const float* __restrict__ user,
                                                  const float* __restrict__ item,
                                                  float* __restrict__ cur,
                                                  float* __restrict__ out) {
    // one float4 per thread over NNODE*16 float4s
    const long long t = (long long)blockIdx.x * 256 + threadIdx.x;
    const int node = (int)(t >> 4);
    const int q    = (int)(t & 15);
    float4 v;
    if (node < N_USERS) v = ((const float4*)user)[(size_t)node * 16 + q];
    else                v = ((const float4*)item)[(size_t)(node - N_USERS) * 16 + q];
    ((float4*)cur)[t] = v;
    // out layout: [node, hop(4), d(64)] -> node row = 64 float4s, hop0 = first 16
    ((float4*)out)[(size_t)node * 64 + q] = v;
}

// ---------------- kernel 2: edge-dropout SpMM (scatter-add) ----------------
__global__ __launch_bounds__(256) void edge_spmm(const float* __restrict__ agg,
                                                 float* __restrict__ nxt,
                                                 const float* __restrict__ vals,
                                                 const int* __restrict__ rows,
                                                 const int* __restrict__ cols,
                                                 uint32_t k0, uint32_t k1) {
    __shared__ float sv[2][256];
    __shared__ int   sr[2][256];
    __shared__ int   sc[2][256];
    const int tid  = threadIdx.x;
    const int lane = tid & 31;
    const int wv   = tid >> 5;                 // 8 waves / block
    const long long base = (long long)blockIdx.x * 1024;   // 4 stages * 256 edges

    // prime stage 0 (each wave stages exactly its own lanes' slots)
    async_g2l_b32(vals + base + tid, &sv[0][tid]);
    async_g2l_b32(rows + base + tid, &sr[0][tid]);
    async_g2l_b32(cols + base + tid, &sc[0][tid]);

    for (int s = 0; s < 4; ++s) {
        const int cur = s & 1;
        if (s + 1 < 4) {
            const long long nb = base + (long long)(s + 1) * 256;
            async_g2l_b32(vals + nb + tid, &sv[cur ^ 1][tid]);
            async_g2l_b32(rows + nb + tid, &sr[cur ^ 1][tid]);
            async_g2l_b32(cols + nb + tid, &sc[cur ^ 1][tid]);
            asm volatile("s_wait_asynccnt 3" ::: "memory"); // current stage's 3 done
        } else {
            asm volatile("s_wait_asynccnt 0" ::: "memory");
        }
        // each wave consumes only its own staged slots -> no barrier needed

        // edge-dropout PRNG: one threefry per lane covers this wave's 32 edges
        const uint32_t ge = (uint32_t)(base + s * 256 + wv * 32 + lane);
        const float    u  = edge_u(k0, k1, ge);
        const uint32_t keepm = (uint32_t)__ballot(u >= 0.5f);  // keep prob 0.5

        #pragma unroll 8
        for (int i = 0; i < 32; ++i) {
            if ((keepm >> i) & 1u) {
                const int   le    = wv * 32 + i;
                const float scale = sv[cur][le] * 2.0f;        // 1/(1-0.5)
                const int   r     = sr[cur][le];
                const int   c     = sc[cur][le];
                const float2 x = *((const float2*)(agg + (size_t)c * DDIM) + lane);
                float* dst = nxt + (size_t)r * DDIM + lane * 2;
                atomic_add_f32_dev(dst,     x.x * scale);
                atomic_add_f32_dev(dst + 1, x.y * scale);
            }
        }
    }
}

// ---------------- kernel 3: message dropout + emit hop slot ----------------
__global__ __launch_bounds__(256) void mess_dropout(const float* __restrict__ nxt,
                                                    float* __restrict__ cur,
                                                    float* __restrict__ out,
                                                    uint32_t k0, uint32_t k1,
                                                    int slot) {
    const uint32_t j = (uint32_t)(blockIdx.x * 256 + threadIdx.x); // [0, HALF_M)
    uint32_t o0, o1;
    threefry2x32(k0, k1, j, j + HALF_M, o0, o1);  // counts pair (j, j+HALF_M)

    // element j  (bits o0) and element j+HALF_M (bits o1)
    {
        const float u = u01_from_bits(o0);
        const float x = nxt[j];
        const float y = (u < 0.9f) ? x * (1.0f / 0.9f) : 0.0f;
        cur[j] = y;
        const uint32_t node = j >> 6, d = j & 63;
        out[((size_t)node * 4 + slot) * DDIM + d] = y;
    }
    {
        const uint32_t idx = j + HALF_M;
        const float u = u01_from_bits(o1);
        const float x = nxt[idx];
        const float y = (u < 0.9f) ? x * (1.0f / 0.9f) : 0.0f;
        cur[idx] = y;
        const uint32_t node = idx >> 6, d = idx & 63;
        out[((size_t)node * 4 + slot) * DDIM + d] = y;
    }
}

// ---------------- host: JAX key derivation ----------------
static void keys_for_hop(int hop, uint32_t& ke0, uint32_t& ke1,
                         uint32_t& km0, uint32_t& km1) {
    // base_key = jax.random.key(42) -> raw (0, 42)
    uint32_t f0, f1;
    threefry2x32(0u, 42u, 0u, (uint32_t)hop, f0, f1);  // fold_in(key, hop)
    // split: bits = threefry(folded, iota(4)); pairs (0,2) and (1,3)
    uint32_t a0, a1, b0, b1;
    threefry2x32(f0, f1, 0u, 2u, a0, a1);
    threefry2x32(f0, f1, 1u, 3u, b0, b1);
    ke0 = a0; ke1 = b0;   // row 0 of reshape(2,2)
    km0 = a1; km1 = b1;   // row 1
}

extern "C" void kernel_launch(void* const* d_in, const int* in_sizes, int n_in,
                              void* d_out, int out_size, void* d_ws, size_t ws_size,
                              hipStream_t stream) {
    (void)in_sizes; (void)n_in; (void)out_size; (void)ws_size;
    const float* user = (const float*)d_in[0];
    const float* item = (const float*)d_in[1];
    const float* vals = (const float*)d_in[2];
    const int*   rows = (const int*)d_in[3];
    const int*   cols = (const int*)d_in[4];
    float* out  = (float*)d_out;

    float* bufA = (float*)d_ws;                 // current embeddings  [N, D]
    float* bufB = bufA + (size_t)NDELEM;        // accumulator         [N, D]

    init_embed<<<NDELEM / 4 / 256, 256, 0, stream>>>(user, item, bufA, out);

    for (int hop = 0; hop < 3; ++hop) {
        uint32_t ke0, ke1, km0, km1;
        keys_for_hop(hop, ke0, ke1, km0, km1);

        hipMemsetAsync(bufB, 0, (size_t)NDELEM * sizeof(float), stream);
        edge_spmm<<<NNZE / 1024, 256, 0, stream>>>(bufA, bufB, vals, rows, cols,
                                                   ke0, ke1);
        mess_dropout<<<HALF_M / 256, 256, 0, stream>>>(bufB, bufA, out,
                                                       km0, km1, hop + 1);
    }
}